// LSTM_30614526886122
// MI455X (gfx1250) — compile-verified
//
#include <hip/hip_runtime.h>

typedef __attribute__((ext_vector_type(2))) float v2f;
typedef __attribute__((ext_vector_type(8))) float v8f;

namespace {
constexpr int B_TOT   = 4096;
constexpr int T_LEN   = 512;
constexpr int D_IN    = 25;
constexpr int H_DIM   = 25;
constexpr int NL      = 3;
constexpr int GATES   = 4 * H_DIM;   // 100
constexpr int KVAL    = 50;          // [in(25) | h(25)] contraction
constexpr int KTOT    = 52;          // padded to multiple of 4 -> 13 K-steps
constexpr int NKK     = 13;
constexpr int TILE_B  = 16;
constexpr int RSTR    = 54;          // LDS row stride (floats); 27 odd -> conflict-free
constexpr int ACTSTR  = 26;
constexpr int THREADS = 128;         // 4 wave32 = 4 gates
}

// sigmoid/tanh built on v_exp_f32 + v_rcp_f32; saturate correctly at +/-inf.
__device__ __forceinline__ float fsigmoid(float v) {
  return __builtin_amdgcn_rcpf(1.0f + __expf(-v));
}
__device__ __forceinline__ float ftanh(float v) {
  return 1.0f - 2.0f * __builtin_amdgcn_rcpf(__expf(2.0f * v) + 1.0f);
}

#if __has_builtin(__builtin_amdgcn_tensor_load_to_lds)
#define USE_TDM 1
typedef unsigned int u32x4 __attribute__((ext_vector_type(4)));
typedef int i32x4 __attribute__((ext_vector_type(4)));
typedef int i32x8 __attribute__((ext_vector_type(8)));

// TDM: DMA one x tile (16 batch rows x 25 f32, batch-row stride T*D elements)
// into LDS at row stride RSTR, using a 25x1 tile iterated 16 times.
// D# packing per CDNA5 ISA.
__device__ __forceinline__ void stage_x_tdm(const float* gsrc, unsigned lds_byte) {
  unsigned long long ga = (unsigned long long)(uintptr_t)gsrc;
  u32x4 g0;
  g0[0] = 1u;                                        // count=1, user-mode D#
  g0[1] = lds_byte;                                  // lds_addr (bytes)
  g0[2] = (unsigned)(ga & 0xffffffffu);              // global_addr[31:0]
  g0[3] = (unsigned)((ga >> 32) & 0x01ffffffu)       // global_addr[56:32]
          | (2u << 30);                              // type = 2 ("image")
  i32x8 g1;
  g1[0] = (2 << 16) | (1 << 19);  // data_size=4B, iterate_enable=1
  g1[1] = (int)(25u << 16);       // tensor_dim0 = 25 (bits 79:48, low half)
  g1[2] = (int)(1u << 16);        // tensor_dim1 = 1  (bits 111:80, low half)
  g1[3] = (int)(25u << 16);       // tile_dim0 = 25   (bits 127:112)
  g1[4] = 1;                      // tile_dim1 = 1 (bits 143:128), tile_dim2 = 0
  g1[5] = 25;                     // tensor_dim0_stride[31:0]
  g1[6] = 0;                      // tensor_dim0_stride[47:32], dim1_stride lo
  g1[7] = 0;
  i32x4 g2;
  g2[0] = 1;                      // tensor_dim2 (inactive)
  g2[1] = RSTR;                   // lds_addr_increment per iteration (elements)
  g2[2] = T_LEN * D_IN;           // global_addr_increment per iteration (elements)
  g2[3] = (int)((TILE_B - 1) << 16);  // iterate_count = 15 -> 16 iterations
  i32x4 g3 = {0, 0, 0, 0};
#if __clang_major__ >= 23
  i32x8 g4 = {0, 0, 0, 0, 0, 0, 0, 0};
  __builtin_amdgcn_tensor_load_to_lds(g0, g1, g2, g3, g4, 0);
#else
  __builtin_amdgcn_tensor_load_to_lds(g0, g1, g2, g3, 0);
#endif
}

__device__ __forceinline__ void wait_tensor0() {
#if __has_builtin(__builtin_amdgcn_s_wait_tensorcnt)
  __builtin_amdgcn_s_wait_tensorcnt(0);
#else
  asm volatile("s_wait_tensorcnt 0x0" ::: "memory");
#endif
}
#else
#define USE_TDM 0
#endif

// Pack weights into WMMA-B layout in d_ws:
// wpad[l][gate][n_local(0..31)][k(0..51)]; k<25 -> Wih, 25<=k<50 -> Whh, else 0.
__global__ void prep_weights(const float* __restrict__ Wih,
                             const float* __restrict__ Whh,
                             float* __restrict__ wpad) {
  int idx = blockIdx.x * blockDim.x + threadIdx.x;
  if (idx >= NL * 4 * 32 * KTOT) return;
  int k    = idx % KTOT;
  int rem  = idx / KTOT;
  int nidx = rem & 31;
  int wg   = (rem >> 5) & 3;
  int l    = rem >> 7;
  int n    = wg * 25 + nidx;
  float v = 0.0f;
  if (nidx < 25) {
    if (k < D_IN)      v = Wih[(l * GATES + n) * D_IN + k];
    else if (k < KVAL) v = Whh[(l * GATES + n) * H_DIM + (k - D_IN)];
  }
  wpad[idx] = v;
}

__launch_bounds__(THREADS)
__global__ void lstm3_fused(const float* __restrict__ x,
                            const float* __restrict__ wpad,
                            const float* __restrict__ bih,
                            const float* __restrict__ bhh,
                            const float* __restrict__ W1,
                            const float* __restrict__ b1,
                            const float* __restrict__ W2,
                            const float* __restrict__ b2,
                            float* __restrict__ out) {
  // Row r = one layer's WMMA-A tile: [input(25) | h(25) | pad2], 16 batch rows.
  // r0 = [xA | h0], r1 = [xB | h0], r2 = [h0 | h1], r3 = [h1 | h2].
  __shared__ float tl[4][TILE_B * RSTR];
  __shared__ float act[4 * TILE_B * ACTSTR];   // activated gates i,f,g,o

  const int tid   = threadIdx.x;
  const int wv    = tid >> 5;      // wave id == gate id (0:i 1:f 2:g 3:o)
  const int ln    = tid & 31;
  const int lhalf = ln >> 4;       // half-wave: K pair select / M+8 rows
  const int lmod  = ln & 15;       // N column within 16-wide tile
  const int b0    = blockIdx.x * TILE_B;

  for (int i = tid; i < 4 * TILE_B * RSTR; i += THREADS)
    (&tl[0][0])[i] = 0.0f;

  // Gate bias splatted per accumulator column (C/D layout: N = lane%16).
  float bias0[NL], bias1[NL];
  #pragma unroll
  for (int l = 0; l < NL; ++l) {
    int n0 = wv * 25 + lmod;
    bias0[l] = bih[l * GATES + n0] + bhh[l * GATES + n0];
    int n1 = wv * 25 + 16 + lmod;
    bias1[l] = (lmod < 9) ? (bih[l * GATES + n1] + bhh[l * GATES + n1]) : 0.0f;
  }
  float c_reg[NL][4] = {};   // cell state, element e = tid + 128*r -> (m,j)

#if USE_TDM
  const unsigned tl_base = (unsigned)(uintptr_t)&tl[0][0];  // LDS byte address
  // Prologue: DMA x_0 into buffer A (row 0); wave 0 owns the TDM.
  if (wv == 0) {
    stage_x_tdm(x + (size_t)b0 * T_LEN * D_IN, tl_base);
    wait_tensor0();
  }
#else
  for (int i = tid; i < TILE_B * D_IN; i += THREADS) {
    int m = i / D_IN, k = i - m * D_IN;
    tl[0][m * RSTR + k] = x[((size_t)(b0 + m) * T_LEN) * D_IN + k];
  }
#endif
  __syncthreads();

  for (int t = 0; t < T_LEN; ++t) {
    // Stage x_{t+1} into the other x buffer; overlaps with this step's WMMAs.
    // End-of-step barrier orders it against step t+1's layer-0 reads.
    if (t + 1 < T_LEN) {
#if USE_TDM
      if (wv == 0)
        stage_x_tdm(x + ((size_t)b0 * T_LEN + (t + 1)) * D_IN,
                    tl_base + (unsigned)(((t + 1) & 1) * TILE_B * RSTR * 4));
#else
      float* xd = tl[(t + 1) & 1];
      for (int i = tid; i < TILE_B * D_IN; i += THREADS) {
        int m = i / D_IN, k = i - m * D_IN;
        xd[m * RSTR + k] = x[((size_t)(b0 + m) * T_LEN + (t + 1)) * D_IN + k];
      }
#endif
    }
    if (t + 2 < T_LEN && tid < TILE_B)
      __builtin_prefetch(&x[((size_t)(b0 + tid) * T_LEN + (t + 2)) * D_IN], 0, 3);

    #pragma unroll
    for (int l = 0; l < NL; ++l) {
      const float* srcA = (l == 0) ? tl[t & 1] : tl[l + 1];
      const float* wb   = wpad + (size_t)((l * 4 + wv) * 32) * KTOT;

      v8f acc0, acc1;
      #pragma unroll
      for (int r = 0; r < 8; ++r) { acc0[r] = bias0[l]; acc1[r] = bias1[l]; }

      // gates[16x32] = A[16x52] @ B[52x32] via 13 K-steps of 16x16x4 f32 WMMA.
      // B fragments are loop-invariant -> compiler keeps them in VGPRs.
      #pragma unroll
      for (int kk = 0; kk < NKK; ++kk) {
        const int k0 = 4 * kk + 2 * lhalf;   // A: lane holds (M=lmod, K=k0,k0+1)
        v2f a   = *(const v2f*)&srcA[lmod * RSTR + k0];
        v2f bf0 = *(const v2f*)&wb[lmod * KTOT + k0];
        v2f bf1 = *(const v2f*)&wb[(16 + lmod) * KTOT + k0];
        acc0 = __builtin_amdgcn_wmma_f32_16x16x4_f32(false, a, false, bf0,
                                                     (short)0, acc0, false, false);
        acc1 = __builtin_amdgcn_wmma_f32_16x16x4_f32(false, a, false, bf1,
                                                     (short)0, acc1, false, false);
      }

      // Activate this wave's gate, stash to LDS (D layout: N=lmod, M=r+8*lhalf).
      const bool is_g = (wv == 2);
      #pragma unroll
      for (int r = 0; r < 8; ++r) {
        int m = r + 8 * lhalf;
        float v0 = is_g ? ftanh(acc0[r]) : fsigmoid(acc0[r]);
        act[(wv * TILE_B + m) * ACTSTR + lmod] = v0;
        if (lmod < 9) {
          float v1 = is_g ? ftanh(acc1[r]) : fsigmoid(acc1[r]);
          act[(wv * TILE_B + m) * ACTSTR + 16 + lmod] = v1;
        }
      }
      __syncthreads();

      // Cross-gate combine: c = sig(f)*c + sig(i)*tanh(g); h = sig(o)*tanh(c).
      // h_l feeds its own tile's h-half and the next layer's input-half.
      #pragma unroll
      for (int r = 0; r < 4; ++r) {
        int e = tid + THREADS * r;
        if (e < TILE_B * H_DIM) {
          int m = e / H_DIM, j = e - m * H_DIM;
          float iv = act[(0 * TILE_B + m) * ACTSTR + j];
          float fv = act[(1 * TILE_B + m) * ACTSTR + j];
          float gv = act[(2 * TILE_B + m) * ACTSTR + j];
          float ov = act[(3 * TILE_B + m) * ACTSTR + j];
          float c  = fv * c_reg[l][r] + iv * gv;
          c_reg[l][r] = c;
          float h = ov * ftanh(c);
          if (l == 0) {
            tl[0][m * RSTR + D_IN + j] = h;
            tl[1][m * RSTR + D_IN + j] = h;
            tl[2][m * RSTR + j]        = h;
          } else if (l == 1) {
            tl[2][m * RSTR + D_IN + j] = h;
            tl[3][m * RSTR + j]        = h;
          } else {
            tl[3][m * RSTR + D_IN + j] = h;
          }
        }
      }
#if USE_TDM
      // Publish the DMA'd x tile before the step-end barrier.
      if (l == NL - 1 && wv == 0) wait_tensor0();
#endif
      __syncthreads();
    }
  }

  // MLP head + softmax on h of layer 2 at t = T-1 (one thread per batch row).
  if (tid < TILE_B) {
    const int m = tid;
    float hloc[H_DIM];
    #pragma unroll
    for (int j = 0; j < H_DIM; ++j) hloc[j] = tl[3][m * RSTR + D_IN + j];
    float z[16];
    for (int u = 0; u < 16; ++u) {
      float s = b1[u];
      for (int j = 0; j < H_DIM; ++j) s += hloc[j] * W1[u * H_DIM + j];
      z[u] = fmaxf(s, 0.0f);
    }
    float lg[14], mx = -3.0e38f;
    for (int v = 0; v < 14; ++v) {
      float s = b2[v];
      for (int u = 0; u < 16; ++u) s += z[u] * W2[v * 16 + u];
      lg[v] = s; mx = fmaxf(mx, s);
    }
    float sum = 0.0f;
    for (int v = 0; v < 14; ++v) { lg[v] = __expf(lg[v] - mx); sum += lg[v]; }
    float inv = 1.0f / sum;
    for (int v = 0; v < 14; ++v) out[(size_t)(b0 + m) * 14 + v] = lg[v] * inv;
  }
}

extern "C" void kernel_launch(void* const* d_in, const int* in_sizes, int n_in,
                              void* d_out, int out_size, void* d_ws, size_t ws_size,
                              hipStream_t stream) {
  const float* x   = (const float*)d_in[0];
  const float* Wih = (const float*)d_in[1];
  const float* Whh = (const float*)d_in[2];
  const float* bih = (const float*)d_in[3];
  const float* bhh = (const float*)d_in[4];
  const float* W1  = (const float*)d_in[5];
  const float* b1  = (const float*)d_in[6];
  const float* W2  = (const float*)d_in[7];
  const float* b2  = (const float*)d_in[8];
  float* out  = (float*)d_out;
  float* wpad = (float*)d_ws;   // 3*4*32*52 floats ~ 80 KB

  const int prep_n = NL * 4 * 32 * KTOT;
  hipLaunchKernelGGL(prep_weights, dim3((prep_n + 255) / 256), dim3(256), 0, stream,
                     Wih, Whh, wpad);
  hipLaunchKernelGGL(lstm3_fused, dim3(B_TOT / TILE_B), dim3(THREADS), 0, stream,
                     x, wpad, bih, bhh, W1, b1, W2, b2, out);
}